// Multi_Intere_Model_7284264534465
// MI455X (gfx1250) — compile-verified
//
#include <hip/hip_runtime.h>
#include <hip/hip_bf16.h>

typedef float v2f __attribute__((ext_vector_type(2)));
typedef float v8f __attribute__((ext_vector_type(8)));

// exact pointee type expected by __builtin_amdgcn_global_load_async_to_lds_b128
typedef int v4i_vs __attribute__((vector_size(4 * sizeof(int))));
typedef __attribute__((address_space(1))) v4i_vs gv4i;   // global (AS1) side
typedef __attribute__((address_space(3))) v4i_vs lv4i;   // LDS   (AS3) side

#define EMBED_DIM   64
#define INTERE_NUM  4
#define SAMPLE_NUM  1024
#define BATCH       1024
#define SEQ_LEN     20
#define NPAIR       ((SEQ_LEN - 1) * BATCH)   // 19456 (step, batch) pairs

// ---------------------------------------------------------------------------
// Kernel A: per-batch attention (ComiRec-SA Intere_build + top-1 routing).
// One wave32 per batch element. Lane t owns dims t and t+32.
// ---------------------------------------------------------------------------
__global__ __launch_bounds__(32) void attn_route_kernel(
    const int* __restrict__ seqs, const float* __restrict__ item,
    const float* __restrict__ Q, float* __restrict__ hitted,
    float* __restrict__ pos) {
  const int b = blockIdx.x;
  const int lane = threadIdx.x;

  __shared__ float seq[SEQ_LEN][EMBED_DIM];   // L2-normalized seq embeddings
  __shared__ float sc[SEQ_LEN][INTERE_NUM];   // scores_full[b]

  float q0[INTERE_NUM], q1[INTERE_NUM];
#pragma unroll
  for (int k = 0; k < INTERE_NUM; ++k) {
    q0[k] = Q[k * EMBED_DIM + lane];
    q1[k] = Q[k * EMBED_DIM + 32 + lane];
  }

  for (int l = 0; l < SEQ_LEN; ++l) {
    const int idx = seqs[b * SEQ_LEN + l];
    const float* row = item + (size_t)idx * EMBED_DIM;
    float e0 = row[lane], e1 = row[32 + lane];
    float ss = e0 * e0 + e1 * e1;
#pragma unroll
    for (int off = 16; off > 0; off >>= 1) ss += __shfl_xor(ss, off, 32);
    const float inv = rsqrtf(ss);
    e0 *= inv; e1 *= inv;
    seq[l][lane] = e0; seq[l][32 + lane] = e1;
#pragma unroll
    for (int k = 0; k < INTERE_NUM; ++k) {
      float d = e0 * q0[k] + e1 * q1[k];
#pragma unroll
      for (int off = 16; off > 0; off >>= 1) d += __shfl_xor(d, off, 32);
      sc[l][k] = d;   // lane-uniform after allreduce; benign same-value write
    }
  }
  __syncthreads();

  for (int i = 1; i < SEQ_LEN; ++i) {
    const float t0 = seq[i][lane], t1 = seq[i][32 + lane];
    float bestdot = -__builtin_inff();
    float h0 = 0.f, h1 = 0.f;
#pragma unroll
    for (int k = 0; k < INTERE_NUM; ++k) {
      float m = -__builtin_inff();
      for (int l = 0; l < i; ++l) m = fmaxf(m, sc[l][k]);
      float denom = 0.f, a0 = 0.f, a1 = 0.f;
      for (int l = 0; l < i; ++l) {
        const float w = __expf(sc[l][k] - m);
        denom += w;
        a0 += w * seq[l][lane];
        a1 += w * seq[l][32 + lane];
      }
      const float r = 1.0f / denom;
      a0 *= r; a1 *= r;
      float d = a0 * t0 + a1 * t1;
#pragma unroll
      for (int off = 16; off > 0; off >>= 1) d += __shfl_xor(d, off, 32);
      if (d > bestdot) { bestdot = d; h0 = a0; h1 = a1; }   // first-max wins
    }
    const size_t o = (size_t)(i - 1) * BATCH + b;
    hitted[o * EMBED_DIM + lane]      = h0;
    hitted[o * EMBED_DIM + 32 + lane] = h1;
    if (lane == 0) pos[o] = bestdot;   // pos_score == dot(target, hitted)
  }
}

// ---------------------------------------------------------------------------
// Kernel B: negative-sample log-sum-exp per (step, batch) pair.
// Gather path: GLOBAL_LOAD_ASYNC_TO_LDS_B128 (ASYNCcnt) -> no VGPR round
// trip, no s_wait_loadcnt stalls in the staging loop.
// Compute path: V_WMMA_F32_16X16X4_F32 pairs per 4-wide K chunk:
//   acc_dot  = A(16 rows) x B(hitted broadcast into 16 cols) -> 16 dots
//   acc_gram = A x A^T (same registers!) -> diagonal = per-row sum-of-squares
// score = dot / ||row||   (normalize-then-dot without materializing norms)
// ---------------------------------------------------------------------------
__device__ inline void lse_combine(float& m, float& s, float om, float os) {
  const float M = fmaxf(m, om);
  if (M == -__builtin_inff()) { m = M; s = 0.f; return; }
  const float a = (m  == -__builtin_inff()) ? 0.f : s  * __expf(m  - M);
  const float c = (om == -__builtin_inff()) ? 0.f : os * __expf(om - M);
  m = M; s = a + c;
}

__device__ __forceinline__ void wait_async0() {
#if __has_builtin(__builtin_amdgcn_s_wait_asynccnt)
  __builtin_amdgcn_s_wait_asynccnt(0);
#else
  asm volatile("s_wait_asynccnt 0x0" ::: "memory");
#endif
}

#define HAVE_ASYNC_LDS \
  (defined(__gfx1250__) && __has_builtin(__builtin_amdgcn_global_load_async_to_lds_b128))

#if defined(__gfx1250__) && __has_builtin(__builtin_amdgcn_global_load_async_to_lds_b128)
template <int OFS>
__device__ __forceinline__ void async_cp_b128(const float* src, float* dst) {
  __builtin_amdgcn_global_load_async_to_lds_b128((gv4i*)src, (lv4i*)dst, OFS,
                                                 /*cpol=*/0);
}
#endif

#define ROWS_BLK   128   // rows staged per pass (16 per wave)
#define ROW_STRIDE 68    // dwords: 272B rows -> 16B aligned + conflict-free frags

__global__ __launch_bounds__(256) void neg_lse_kernel(
    const int* __restrict__ neg, const float* __restrict__ item,
    const float* __restrict__ hitted, const float* __restrict__ pos,
    float* __restrict__ partial) {
  const int pair = blockIdx.x;            // (i-1)*BATCH + b
  const int t = threadIdx.x;
  const int lane = t & 31;
  const int wave = t >> 5;

  __shared__ float lds_rows[ROWS_BLK * ROW_STRIDE];   // ~34 KB staging
  __shared__ float lds_h[EMBED_DIM];
  __shared__ float red_m[8], red_s[8];

  if (t < EMBED_DIM) lds_h[t] = hitted[(size_t)pair * EMBED_DIM + t];

  float run_m = -__builtin_inff();
  float run_s = 0.f;

  const long long sbase = (long long)pair * SAMPLE_NUM;
  const int r = t >> 1;          // row 0..127 this thread gathers
  const int half = t & 1;        // which 32-float half of the row

  for (int st = 0; st < SAMPLE_NUM / ROWS_BLK; ++st) {
    // -------- gather 128 rows: 8 x async B128 per thread, memory -> LDS
    const int idx = neg[sbase + st * ROWS_BLK + r];
    const float* srcp = item + (size_t)idx * EMBED_DIM + half * 32;
    float* dstp = &lds_rows[r * ROW_STRIDE + half * 32];
#if defined(__gfx1250__) && __has_builtin(__builtin_amdgcn_global_load_async_to_lds_b128)
    async_cp_b128<0>(srcp, dstp);
    async_cp_b128<16>(srcp, dstp);
    async_cp_b128<32>(srcp, dstp);
    async_cp_b128<48>(srcp, dstp);
    async_cp_b128<64>(srcp, dstp);
    async_cp_b128<80>(srcp, dstp);
    async_cp_b128<96>(srcp, dstp);
    async_cp_b128<112>(srcp, dstp);
    wait_async0();
#else
    const float4* src4 = (const float4*)srcp;
#pragma unroll
    for (int q = 0; q < 8; ++q) {
      const float4 v = src4[q];
      dstp[q * 4 + 0] = v.x; dstp[q * 4 + 1] = v.y;
      dstp[q * 4 + 2] = v.z; dstp[q * 4 + 3] = v.w;
    }
#endif
    __syncthreads();

    // -------- WMMA: this wave scores rows 16*wave .. 16*wave+15
    v8f acc_dot  = {0.f, 0.f, 0.f, 0.f, 0.f, 0.f, 0.f, 0.f};
    v8f acc_gram = {0.f, 0.f, 0.f, 0.f, 0.f, 0.f, 0.f, 0.f};
    const int mrow = 16 * wave + (lane & 15);
    const int koff = (lane < 16) ? 0 : 2;   // A layout: lanes 0-15: K0,K1; 16-31: K2,K3
    const float* arow = &lds_rows[mrow * ROW_STRIDE];
#pragma unroll
    for (int c = 0; c < EMBED_DIM / 4; ++c) {
      v2f a, bf;
      a.x  = arow[4 * c + koff];
      a.y  = arow[4 * c + koff + 1];
      bf.x = lds_h[4 * c + koff];           // hitted broadcast into all 16 cols
      bf.y = lds_h[4 * c + koff + 1];
      acc_dot  = __builtin_amdgcn_wmma_f32_16x16x4_f32(false, a, false, bf,
                                                       (short)0, acc_dot, false, false);
      acc_gram = __builtin_amdgcn_wmma_f32_16x16x4_f32(false, a, false, a,
                                                       (short)0, acc_gram, false, false);
    }

    // -------- extract: Gram diagonal lives at canonical lanes
    // {0..7} (rows 0..7, vgpr=lane) and {24..31} (rows 8..15, vgpr=lane-24)
    const bool valid = (lane < 8) || (lane >= 24);
    const int vidx = (lane < 16) ? (lane & 7) : ((lane - 24) & 7);
    float dotv = acc_dot[0], ssv = acc_gram[0];
#pragma unroll
    for (int j = 1; j < 8; ++j) {
      if (vidx == j) { dotv = acc_dot[j]; ssv = acc_gram[j]; }
    }
    const float score = valid ? dotv * rsqrtf(ssv) : -__builtin_inff();

    // -------- online log-sum-exp (per lane)
    if (score > run_m) {
      run_s = run_s * __expf(run_m - score) + 1.0f;
      run_m = score;
    } else if (valid) {
      run_s += __expf(score - run_m);
    }
    __syncthreads();   // protect lds_rows before next gather pass
  }

  // -------- wave reduction of (m, s)
#pragma unroll
  for (int off = 16; off > 0; off >>= 1) {
    const float om = __shfl_xor(run_m, off, 32);
    const float os = __shfl_xor(run_s, off, 32);
    lse_combine(run_m, run_s, om, os);
  }
  if (lane == 0) { red_m[wave] = run_m; red_s[wave] = run_s; }
  __syncthreads();
  if (t == 0) {
    float m = red_m[0], s = red_s[0];
#pragma unroll
    for (int w = 1; w < 8; ++w) lse_combine(m, s, red_m[w], red_s[w]);
    const float lse = m + __logf(s);
    partial[pair] = lse - pos[pair];
  }
}

// ---------------------------------------------------------------------------
// Kernel C: deterministic fixed-tree reduction (no float atomics => bit-stable
// across graph replays).
// ---------------------------------------------------------------------------
__global__ __launch_bounds__(256) void reduce_kernel(
    const float* __restrict__ partial, float* __restrict__ out, int n) {
  __shared__ float buf[256];
  const int t = threadIdx.x;
  float s = 0.f;
  for (int i = t; i < n; i += 256) s += partial[i];
  buf[t] = s;
  __syncthreads();
  for (int off = 128; off > 0; off >>= 1) {
    if (t < off) buf[t] += buf[t + off];
    __syncthreads();
  }
  if (t == 0) out[0] = buf[0];
}

// ---------------------------------------------------------------------------
extern "C" void kernel_launch(void* const* d_in, const int* in_sizes, int n_in,
                              void* d_out, int out_size, void* d_ws, size_t ws_size,
                              hipStream_t stream) {
  const int*   seqs = (const int*)d_in[0];     // [1024, 20]
  const int*   neg  = (const int*)d_in[1];     // [19, 1024, 1024]
  const float* item = (const float*)d_in[2];   // [1000000, 64]
  const float* Q    = (const float*)d_in[3];   // [4, 64]
  float* out = (float*)d_out;                  // scalar f32 loss

  float* hitted  = (float*)d_ws;                           // NPAIR * 64 floats
  float* pos     = hitted + (size_t)NPAIR * EMBED_DIM;     // NPAIR floats
  float* partial = pos + NPAIR;                            // NPAIR floats

  attn_route_kernel<<<BATCH, 32, 0, stream>>>(seqs, item, Q, hitted, pos);
  neg_lse_kernel<<<NPAIR, 256, 0, stream>>>(neg, item, hitted, pos, partial);
  reduce_kernel<<<1, 256, 0, stream>>>(partial, out, NPAIR);
}